// APPNP_11991548690476
// MI455X (gfx1250) — compile-verified
//
#include <hip/hip_runtime.h>

typedef __attribute__((ext_vector_type(2))) float v2f;
typedef __attribute__((ext_vector_type(8))) float v8f;

#define C_IN   512
#define C_HID  256
#define C_OUT  64
#define ALPHA  0.1f
#define KSTEPS 10

#define XS_STRIDE 516   // 512 + 4 pad: lane m -> bank (m*4)%64, conflict-free
#define HS_STRIDE 260   // 256 + 4 pad

// ---------------- degree count / normalization ----------------
__global__ void k_cnt_init(int* cnt, int N) {
    int i = blockIdx.x * blockDim.x + threadIdx.x;
    if (i < N) cnt[i] = 0;
}

__global__ void k_cnt_acc(const long long* __restrict__ ei, int* cnt, int E) {
    int e = blockIdx.x * blockDim.x + threadIdx.x;
    if (e < E) {
        int d = (int)ei[(size_t)E + e];   // dst row
        atomicAdd(&cnt[d], 1);
    }
}

__global__ void k_dinv(const int* __restrict__ cnt, float* dinv, int N) {
    int i = blockIdx.x * blockDim.x + threadIdx.x;
    if (i < N) dinv[i] = rsqrtf((float)cnt[i] + 1.0f);   // +1 self-loop
}

// ---------------- exclusive scan (rowptr) ----------------
__global__ __launch_bounds__(256) void k_scan1(const int* __restrict__ cnt,
                                               int* __restrict__ rowptr,
                                               int* __restrict__ bsum, int N)
{
    __shared__ int sh[256];
    int tid = threadIdx.x;
    int i = blockIdx.x * 256 + tid;
    int x = (i < N) ? cnt[i] : 0;
    sh[tid] = x;
    __syncthreads();
    for (int off = 1; off < 256; off <<= 1) {   // Hillis-Steele inclusive
        int v = (tid >= off) ? sh[tid - off] : 0;
        __syncthreads();
        sh[tid] += v;
        __syncthreads();
    }
    if (i < N) rowptr[i] = sh[tid] - x;         // exclusive within block
    if (tid == 255) bsum[blockIdx.x] = sh[255];
}

__global__ void k_scan2(int* bsum, int nb) {    // tiny one-time serial pass
    if (blockIdx.x == 0 && threadIdx.x == 0) {
        int run = 0;
        for (int i = 0; i < nb; ++i) { int v = bsum[i]; bsum[i] = run; run += v; }
    }
}

__global__ __launch_bounds__(256) void k_scan3(int* rowptr, const int* __restrict__ bsum,
                                               int N, int E)
{
    int i = blockIdx.x * 256 + threadIdx.x;
    if (i < N) rowptr[i] += bsum[i >> 8];
    if (i == 0) rowptr[N] = E;
}

__global__ void k_pos_copy(const int* __restrict__ rowptr, int* pos, int N) {
    int i = blockIdx.x * blockDim.x + threadIdx.x;
    if (i < N) pos[i] = rowptr[i];
}

// CSR fill: bucket edges by dst, precompute (1-a)*dinv[s]*dinv[d]
__global__ void k_fill(const long long* __restrict__ ei, const float* __restrict__ dinv,
                       int* pos, int* __restrict__ csrc, float* __restrict__ cw, int E)
{
    int e = blockIdx.x * blockDim.x + threadIdx.x;
    if (e < E) {
        int s = (int)ei[e];
        int d = (int)ei[(size_t)E + e];
        int slot = atomicAdd(&pos[d], 1);
        csrc[slot] = s;
        cw[slot]   = (1.0f - ALPHA) * dinv[s] * dinv[d];
    }
}

// ---------------- fused MLP: h = relu(x@W1+b1)@W2 + b2 ----------------
// One block = 16 rows, 4 waves (128 threads). fp32 WMMA 16x16x4.
__global__ __launch_bounds__(128) void k_mlp(
    const float* __restrict__ x,  const float* __restrict__ W1,
    const float* __restrict__ b1, const float* __restrict__ W2,
    const float* __restrict__ b2, float* __restrict__ h, int N)
{
    __shared__ float Xs[16 * XS_STRIDE];
    __shared__ float Hs[16 * HS_STRIDE];

    const int tid = threadIdx.x;
    const int w   = tid >> 5;        // wave 0..3
    const int t   = tid & 31;        // lane
    const int m   = t & 15;          // row/col within frag
    const int g   = t >> 4;          // lane half (K pairing)
    const int r0  = blockIdx.x * 16;

    // stage 0: load 16x512 x-tile to LDS (float4, coalesced)
    for (int i = tid; i < 16 * 128; i += 128) {
        int row = i >> 7;
        int c4  = (i & 127) << 2;
        float4 v = make_float4(0.f, 0.f, 0.f, 0.f);
        if (r0 + row < N) v = *(const float4*)&x[(size_t)(r0 + row) * C_IN + c4];
        *(float4*)&Xs[row * XS_STRIDE + c4] = v;
    }
    __syncthreads();

    // stage 1: hidden[16][256] = relu(Xtile @ W1 + b1); wave w owns cols [w*64, w*64+64)
    {
        v8f acc[4] = {};
        const int nbase = w * 64 + m;
        for (int k0 = 0; k0 < C_IN; k0 += 4) {
            v2f a;
            a[0] = Xs[m * XS_STRIDE + k0 + 2 * g];
            a[1] = Xs[m * XS_STRIDE + k0 + 2 * g + 1];
            const float* wp = &W1[(size_t)(k0 + 2 * g) * C_HID + nbase];
#pragma unroll
            for (int nt = 0; nt < 4; ++nt) {
                v2f b;
                b[0] = wp[nt * 16];
                b[1] = wp[C_HID + nt * 16];
                acc[nt] = __builtin_amdgcn_wmma_f32_16x16x4_f32(
                    false, a, false, b, (short)0, acc[nt], false, false);
            }
        }
#pragma unroll
        for (int nt = 0; nt < 4; ++nt) {
            int nc = w * 64 + nt * 16 + m;
            float bb = b1[nc];
#pragma unroll
            for (int v = 0; v < 8; ++v) {      // D: row = v + 8*g, col = m
                float val = acc[nt][v] + bb;
                Hs[(v + 8 * g) * HS_STRIDE + nc] = val > 0.f ? val : 0.f;
            }
        }
    }
    __syncthreads();

    // stage 2: out[16][64] = hidden @ W2 + b2; wave w owns cols [w*16, w*16+16)
    {
        v8f acc = {};
        const int n0 = w * 16;
        for (int k0 = 0; k0 < C_HID; k0 += 4) {
            v2f a;
            a[0] = Hs[m * HS_STRIDE + k0 + 2 * g];
            a[1] = Hs[m * HS_STRIDE + k0 + 2 * g + 1];
            const float* wp = &W2[(size_t)(k0 + 2 * g) * C_OUT + n0 + m];
            v2f b;
            b[0] = wp[0];
            b[1] = wp[C_OUT];
            acc = __builtin_amdgcn_wmma_f32_16x16x4_f32(
                false, a, false, b, (short)0, acc, false, false);
        }
        float bb = b2[n0 + m];
#pragma unroll
        for (int v = 0; v < 8; ++v) {
            int row = r0 + v + 8 * g;
            if (row < N) h[(size_t)row * C_OUT + n0 + m] = acc[v] + bb;
        }
    }
}

// ---------------- APPNP propagation: pure gather, one wave32 per node ----------------
// nxt[i] = (1-a) * sum_{e: dst=i} norm_e * cur[src_e]  +  (1-a)*dinv[i]^2*cur[i] + a*h[i]
// lane l owns channels {2l, 2l+1}: one coalesced b64 row-gather per edge.
__global__ __launch_bounds__(256) void k_prop_gather(
    const int* __restrict__ rowptr, const int* __restrict__ csrc,
    const float* __restrict__ cw,   const float* __restrict__ dinv,
    const float* __restrict__ cur,  const float* __restrict__ h,
    float* __restrict__ nxt, int N)
{
    int gid  = blockIdx.x * blockDim.x + threadIdx.x;
    int node = gid >> 5;
    int lane = gid & 31;
    if (node >= N) return;

    const float2* cur2 = (const float2*)cur;
    float  di = dinv[node];
    float  sw = (1.0f - ALPHA) * di * di;
    float2 c  = cur2[(size_t)node * 32 + lane];
    float2 hv = ((const float2*)h)[(size_t)node * 32 + lane];
    float a0 = sw * c.x + ALPHA * hv.x;
    float a1 = sw * c.y + ALPHA * hv.y;

    int jb = rowptr[node];
    int je = rowptr[node + 1];
    for (int j = jb; j < je; ++j) {
        int   s = __builtin_amdgcn_readfirstlane(csrc[j]);  // wave-uniform -> saddr gather
        float w = cw[j];
        float2 v = cur2[(size_t)s * 32 + lane];
        a0 = fmaf(w, v.x, a0);
        a1 = fmaf(w, v.y, a1);
    }
    float2 o; o.x = a0; o.y = a1;
    ((float2*)nxt)[(size_t)node * 32 + lane] = o;
}

// ---------------- log_softmax over 64 channels, one wave32 per node ----------------
__global__ void k_logsoftmax(float* __restrict__ out, int N) {
    int gid  = blockIdx.x * blockDim.x + threadIdx.x;
    int node = gid >> 5;
    int lane = gid & 31;
    if (node >= N) return;
    float v0 = out[(size_t)node * C_OUT + lane];
    float v1 = out[(size_t)node * C_OUT + 32 + lane];
    float mx = fmaxf(v0, v1);
#pragma unroll
    for (int off = 16; off > 0; off >>= 1)
        mx = fmaxf(mx, __shfl_xor(mx, off, 32));
    float s = expf(v0 - mx) + expf(v1 - mx);
#pragma unroll
    for (int off = 16; off > 0; off >>= 1)
        s += __shfl_xor(s, off, 32);
    float l = mx + logf(s);
    out[(size_t)node * C_OUT + lane]      = v0 - l;
    out[(size_t)node * C_OUT + 32 + lane] = v1 - l;
}

// ---------------- launch ----------------
extern "C" void kernel_launch(void* const* d_in, const int* in_sizes, int n_in,
                              void* d_out, int out_size, void* d_ws, size_t ws_size,
                              hipStream_t stream)
{
    const float*     x  = (const float*)d_in[0];
    const long long* ei = (const long long*)d_in[1];   // int64 [2, E]
    const float*     W1 = (const float*)d_in[2];
    const float*     b1 = (const float*)d_in[3];
    const float*     W2 = (const float*)d_in[4];
    const float*     b2 = (const float*)d_in[5];

    const int N  = in_sizes[0] / C_IN;
    const int E  = in_sizes[1] / 2;
    const int nb = (N + 255) / 256;

    // workspace layout (8-byte alignment preserved for float2 rows)
    float* ws     = (float*)d_ws;
    float* h      = ws;                                  // [64N] f32
    float* outA   = h    + (size_t)N * C_OUT;            // [64N] f32
    float* dinv   = outA + (size_t)N * C_OUT;            // [N]   f32
    float* cw     = dinv + N;                            // [E]   f32  (sorted-by-dst weights)
    int*   cnt    = (int*)(cw + E);                      // [N]   i32  (later reused as pos)
    int*   rowptr = cnt + N;                             // [N+1] i32
    int*   csrc   = rowptr + (N + 1);                    // [E]   i32  (sorted-by-dst src ids)
    int*   bsum   = csrc + E;                            // [nb]  i32
    int*   pos    = cnt;                                 // reuse after scan

    const int T = 256;
    // --- degree + CSR build (one-time, integer atomics only) ---
    k_cnt_init<<<(N + T - 1) / T, T, 0, stream>>>(cnt, N);
    k_cnt_acc <<<(E + T - 1) / T, T, 0, stream>>>(ei, cnt, E);
    k_dinv    <<<(N + T - 1) / T, T, 0, stream>>>(cnt, dinv, N);
    k_scan1   <<<nb, 256, 0, stream>>>(cnt, rowptr, bsum, N);
    k_scan2   <<<1, 32, 0, stream>>>(bsum, nb);
    k_scan3   <<<nb, 256, 0, stream>>>(rowptr, bsum, N, E);
    k_pos_copy<<<(N + T - 1) / T, T, 0, stream>>>(rowptr, pos, N);
    k_fill    <<<(E + T - 1) / T, T, 0, stream>>>(ei, dinv, pos, csrc, cw, E);

    // --- MLP (fp32 WMMA) ---
    k_mlp<<<(N + 15) / 16, 128, 0, stream>>>(x, W1, b1, W2, b2, h, N);

    // --- K-step propagation, atomic-free gather, ping-pong ending in d_out ---
    const float* cur = h;
    for (int s = 0; s < KSTEPS; ++s) {
        float* nxt = (s % 2 == 0) ? outA : (float*)d_out;
        k_prop_gather<<<((size_t)N * 32 + T - 1) / T, T, 0, stream>>>(
            rowptr, csrc, cw, dinv, cur, h, nxt, N);
        cur = nxt;
    }
    // after 10 steps cur == d_out
    k_logsoftmax<<<((size_t)N * 32 + T - 1) / T, T, 0, stream>>>((float*)d_out, N);
}